// HeteroGraphConv_8211977470437
// MI455X (gfx1250) — compile-verified
//
#include <hip/hip_runtime.h>

#define D   128
#define EPS 1e-5f
#define PS  288   // LDS dwords per k-pair row: 128 cols * 2 + 32 pad (PS%64==32)
#define HS  132   // LDS row stride (dwords) for per-wave h1 staging (bank-pad)

typedef float v2f __attribute__((ext_vector_type(2)));
typedef float v8f __attribute__((ext_vector_type(8)));

typedef __attribute__((address_space(1))) int* global_i_p;
typedef __attribute__((address_space(3))) int* shared_i_p;

#if defined(__has_builtin)
#if __has_builtin(__builtin_amdgcn_global_load_async_to_lds_b32)
#define HAVE_ASYNC_LDS 1
#endif
#endif
#ifndef HAVE_ASYNC_LDS
#define HAVE_ASYNC_LDS 0
#endif

__device__ __forceinline__ v8f wmma_f32(v2f a, v2f b, v8f c) {
  // D = A(16x4 f32) * B(4x16 f32) + C(16x16 f32), exact f32 accumulate
  return __builtin_amdgcn_wmma_f32_16x16x4_f32(false, a, false, b, (short)0, c,
                                               false, false);
}

// Stage a 128x128 f32 weight matrix into LDS, k-pair interleaved:
//   lds[(k>>1)*PS + c*2 + (k&1)] = W[k][c]
// so a WMMA B-fragment (W[k0][c], W[k0+1][c]) is ONE aligned ds_load_b64.
// Global side: 32 consecutive lanes read 128 contiguous bytes (coalesced).
// LDS side: banks 2*lane (mod 64) -> conflict-free.
__device__ __forceinline__ void stage_weight(const float* __restrict__ W,
                                             float* __restrict__ lds, int tid) {
#pragma unroll
  for (int i = 0; i < 64; ++i) {
    const int f = i * 256 + tid;  // dword index 0..16383
    const int k = f >> 7;
    const int c = f & 127;
    const int laddr = (k >> 1) * PS + c * 2 + (k & 1);
#if HAVE_ASYNC_LDS
    __builtin_amdgcn_global_load_async_to_lds_b32(
        (global_i_p)(W + f), (shared_i_p)(lds + laddr), 0, 0);
#else
    lds[laddr] = W[f];
#endif
  }
}

__device__ __forceinline__ void stage_commit() {
#if HAVE_ASYNC_LDS
#if __has_builtin(__builtin_amdgcn_s_wait_asynccnt)
  __builtin_amdgcn_s_wait_asynccnt(0);
#else
  asm volatile("s_wait_asynccnt 0x0" ::: "memory");
#endif
#endif
  __syncthreads();
}

// B-fragment from pair-interleaved LDS weights: one aligned ds_load_b64.
__device__ __forceinline__ v2f bfrag(const float* __restrict__ wl, int p, int c) {
  return *(const v2f*)(wl + p * PS + c * 2);
}

// ---------------------------------------------------------------------------
// Edge MLP: T = relu(X @ W1 + b1) @ W2 + b2
// 256 threads = 8 waves; each wave computes a 16x128 output strip.
// ---------------------------------------------------------------------------
__global__ __launch_bounds__(256) void edge_mlp_kernel(
    const float* __restrict__ X,
    const float* __restrict__ W1, const float* __restrict__ B1,
    const float* __restrict__ W2, const float* __restrict__ B2,
    float* __restrict__ T, int N)
{
  __shared__ float wlds[64 * PS];         // 72 KB paired weight matrix
  __shared__ float hstage[8 * 16 * HS];   // 66 KB per-wave relu(h1)

  const int tid   = threadIdx.x;
  const int lane  = tid & 31;
  const int wave  = tid >> 5;
  const int mrow  = lane & 15;   // M (A) / N (B,C) index
  const int khalf = lane >> 4;   // K-pair select (A/B) / M+8 rows (C/D)
  const int mbase = blockIdx.x * 128 + wave * 16;

  float* hs = hstage + wave * (16 * HS);

  stage_weight(W1, wlds, tid);
  stage_commit();

  // ---- GEMM1: h1 = X @ W1 + b1 ----
  v8f acc[8];
#pragma unroll
  for (int t = 0; t < 8; ++t) {
    const float bv = B1[t * 16 + mrow];
#pragma unroll
    for (int v = 0; v < 8; ++v) acc[t][v] = bv;
  }

  int arow = mbase + mrow;
  if (arow >= N) arow = N - 1;  // clamp; tail rows never stored
  const float* __restrict__ xrow = X + (size_t)arow * D;

#pragma unroll 4
  for (int kb = 0; kb < D / 4; ++kb) {
    const int k0 = kb * 4 + khalf * 2;
    const int p  = kb * 2 + khalf;
    const v2f a = *(const v2f*)(xrow + k0);
#pragma unroll
    for (int t = 0; t < 8; ++t)
      acc[t] = wmma_f32(a, bfrag(wlds, p, t * 16 + mrow), acc[t]);
  }

  // ReLU, stage to LDS (padded rows -> conflict-free A-fragment re-reads)
#pragma unroll
  for (int t = 0; t < 8; ++t) {
#pragma unroll
    for (int v = 0; v < 8; ++v)
      hs[(khalf * 8 + v) * HS + t * 16 + mrow] = fmaxf(acc[t][v], 0.0f);
  }

  // Re-stage weights: W2 overwrites W1 (all waves must be done reading W1)
  __syncthreads();
  stage_weight(W2, wlds, tid);
  stage_commit();

  // ---- GEMM2: t = h1 @ W2 + b2 ----
#pragma unroll
  for (int t = 0; t < 8; ++t) {
    const float bv = B2[t * 16 + mrow];
#pragma unroll
    for (int v = 0; v < 8; ++v) acc[t][v] = bv;
  }

#pragma unroll 4
  for (int kb = 0; kb < D / 4; ++kb) {
    const int k0 = kb * 4 + khalf * 2;
    const int p  = kb * 2 + khalf;
    const v2f a = *(const v2f*)(hs + mrow * HS + k0);  // ds_load_b64
#pragma unroll
    for (int t = 0; t < 8; ++t)
      acc[t] = wmma_f32(a, bfrag(wlds, p, t * 16 + mrow), acc[t]);
  }

#pragma unroll
  for (int t = 0; t < 8; ++t) {
#pragma unroll
    for (int v = 0; v < 8; ++v) {
      const int r = mbase + khalf * 8 + v;
      if (r < N) T[(size_t)r * D + t * 16 + mrow] = acc[t][v];
    }
  }
}

// ---------------------------------------------------------------------------
// Gather + per-edge scale + scatter-add. One wave per edge, float4 per lane.
// Source rows and destination rows are L2-resident (25.6 MB vs 192 MB L2).
// ---------------------------------------------------------------------------
__global__ __launch_bounds__(256) void scatter_kernel(
    const float* __restrict__ T, const int* __restrict__ ei,
    const float* __restrict__ ea, float* __restrict__ aggr, int E)
{
  const int lane = threadIdx.x & 31;
  const int e    = blockIdx.x * 8 + (threadIdx.x >> 5);
  if (e >= E) return;

  const int   src = ei[e];        // ei[0][e]
  const int   dst = ei[E + e];    // ei[1][e]
  const float w   = ea[e];

  const float4 v = ((const float4*)(T + (size_t)src * D))[lane];
  float* ad = aggr + (size_t)dst * D + lane * 4;
  atomicAdd(ad + 0, v.x * w);
  atomicAdd(ad + 1, v.y * w);
  atomicAdd(ad + 2, v.z * w);
  atomicAdd(ad + 3, v.w * w);
}

// ---------------------------------------------------------------------------
// Node update: OUT = relu(LayerNorm(A @ WU + bu) * g + be)
// WMMA strip layout; LayerNorm in registers + half-wave shuffle reduction.
// ---------------------------------------------------------------------------
__global__ __launch_bounds__(256) void node_update_kernel(
    const float* __restrict__ A,
    const float* __restrict__ WU, const float* __restrict__ BU,
    const float* __restrict__ G,  const float* __restrict__ BE,
    float* __restrict__ OUT, int N)
{
  __shared__ float wlds[64 * PS];  // 72 KB paired weight matrix

  const int tid   = threadIdx.x;
  const int lane  = tid & 31;
  const int wave  = tid >> 5;
  const int mrow  = lane & 15;
  const int khalf = lane >> 4;
  const int mbase = blockIdx.x * 128 + wave * 16;

  stage_weight(WU, wlds, tid);
  stage_commit();

  v8f acc[8];
#pragma unroll
  for (int t = 0; t < 8; ++t) {
    const float bv = BU[t * 16 + mrow];
#pragma unroll
    for (int v = 0; v < 8; ++v) acc[t][v] = bv;
  }

  int arow = mbase + mrow;
  if (arow >= N) arow = N - 1;
  const float* __restrict__ xrow = A + (size_t)arow * D;

#pragma unroll 4
  for (int kb = 0; kb < D / 4; ++kb) {
    const int k0 = kb * 4 + khalf * 2;
    const int p  = kb * 2 + khalf;
    const v2f a = *(const v2f*)(xrow + k0);
#pragma unroll
    for (int t = 0; t < 8; ++t)
      acc[t] = wmma_f32(a, bfrag(wlds, p, t * 16 + mrow), acc[t]);
  }

  // Row statistics: each lane holds 8 rows x 8 cols of the 16x128 strip.
  float mu[8], inv[8];
#pragma unroll
  for (int v = 0; v < 8; ++v) {
    float s1 = 0.0f, s2 = 0.0f;
#pragma unroll
    for (int t = 0; t < 8; ++t) {
      const float h = acc[t][v];
      s1 += h;
      s2 += h * h;
    }
    // reduce over the 16 lanes of this half-wave (masks < 16 keep bit 4)
#pragma unroll
    for (int m = 8; m >= 1; m >>= 1) {
      s1 += __shfl_xor(s1, m, 32);
      s2 += __shfl_xor(s2, m, 32);
    }
    const float mean = s1 * (1.0f / D);
    const float var  = s2 * (1.0f / D) - mean * mean;
    mu[v]  = mean;
    inv[v] = rsqrtf(var + EPS);
  }

#pragma unroll
  for (int t = 0; t < 8; ++t) {
    const int   c  = t * 16 + mrow;
    const float gv = G[c];
    const float bv = BE[c];
#pragma unroll
    for (int v = 0; v < 8; ++v) {
      const int r = mbase + khalf * 8 + v;
      if (r < N) {
        const float h = (acc[t][v] - mu[v]) * inv[v] * gv + bv;
        OUT[(size_t)r * D + c] = fmaxf(h, 0.0f);
      }
    }
  }
}

// ---------------------------------------------------------------------------
extern "C" void kernel_launch(void* const* d_in, const int* in_sizes, int n_in,
                              void* d_out, int out_size, void* d_ws, size_t ws_size,
                              hipStream_t stream)
{
  const float* x_a   = (const float*)d_in[0];
  const float* x_b   = (const float*)d_in[1];
  const int*   ei_ab = (const int*)  d_in[2];
  const float* ea_ab = (const float*)d_in[3];
  const int*   ei_ba = (const int*)  d_in[4];
  const float* ea_ba = (const float*)d_in[5];
  const float* w1_ab = (const float*)d_in[6];
  const float* b1_ab = (const float*)d_in[7];
  const float* w2_ab = (const float*)d_in[8];
  const float* b2_ab = (const float*)d_in[9];
  const float* w1_ba = (const float*)d_in[10];
  const float* b1_ba = (const float*)d_in[11];
  const float* w2_ba = (const float*)d_in[12];
  const float* b2_ba = (const float*)d_in[13];
  const float* wu_a  = (const float*)d_in[14];
  const float* bu_a  = (const float*)d_in[15];
  const float* g_a   = (const float*)d_in[16];
  const float* be_a  = (const float*)d_in[17];
  const float* wu_b  = (const float*)d_in[18];
  const float* bu_b  = (const float*)d_in[19];
  const float* g_b   = (const float*)d_in[20];
  const float* be_b  = (const float*)d_in[21];

  const int NA   = in_sizes[0] / D;
  const int NB   = in_sizes[1] / D;
  const int E_ab = in_sizes[3];
  const int E_ba = in_sizes[5];

  // Workspace layout (all f32): t_ab | t_ba | aggr_a | aggr_b
  float* t_ab   = (float*)d_ws;
  float* t_ba   = t_ab   + (size_t)NA * D;
  float* aggr_a = t_ba   + (size_t)NB * D;
  float* aggr_b = aggr_a + (size_t)NA * D;

  // Residual init: aggr = x (scatter adds on top)
  (void)hipMemcpyAsync(aggr_a, x_a, (size_t)NA * D * sizeof(float),
                       hipMemcpyDeviceToDevice, stream);
  (void)hipMemcpyAsync(aggr_b, x_b, (size_t)NB * D * sizeof(float),
                       hipMemcpyDeviceToDevice, stream);

  const dim3 blk(256);
  const int gA = (NA + 127) / 128;
  const int gB = (NB + 127) / 128;

  edge_mlp_kernel<<<gA, blk, 0, stream>>>(x_a, w1_ab, b1_ab, w2_ab, b2_ab, t_ab, NA);
  edge_mlp_kernel<<<gB, blk, 0, stream>>>(x_b, w1_ba, b1_ba, w2_ba, b2_ba, t_ba, NB);

  scatter_kernel<<<(E_ab + 7) / 8, blk, 0, stream>>>(t_ab, ei_ab, ea_ab, aggr_b, E_ab);
  scatter_kernel<<<(E_ba + 7) / 8, blk, 0, stream>>>(t_ba, ei_ba, ea_ba, aggr_a, E_ba);

  float* out_a = (float*)d_out;
  float* out_b = out_a + (size_t)NA * D;
  node_update_kernel<<<gA, blk, 0, stream>>>(aggr_a, wu_a, bu_a, g_a, be_a, out_a, NA);
  node_update_kernel<<<gB, blk, 0, stream>>>(aggr_b, wu_b, bu_b, g_b, be_b, out_b, NB);
}